// GCNBlock_35871566856216
// MI455X (gfx1250) — compile-verified
//
#include <hip/hip_runtime.h>

// GCN block for MI455X (gfx1250, wave32).
// Per layer: h = x@W via V_WMMA_F32_16X16X4_F32 (fp32 matrix path),
// then L2-resident gather/scatter-atomic aggregation over edges,
// then fused bias(+ReLU). Scatter dominates; h/agg (25.6MB each) live in L2.

typedef float v2f __attribute__((ext_vector_type(2)));
typedef float v8f __attribute__((ext_vector_type(8)));

#define FDIM 64

__global__ void zero_f32_kernel(float* __restrict__ p, int n) {
  int i = blockIdx.x * blockDim.x + threadIdx.x;
  if (i < n) p[i] = 0.0f;
}

// deg[dst[e]] += 1 for every edge (hardware f32 atomic add).
__global__ void degree_kernel(const int* __restrict__ dst, float* __restrict__ deg, int E) {
  int e = blockIdx.x * blockDim.x + threadIdx.x;
  if (e < E) unsafeAtomicAdd(&deg[dst[e]], 1.0f);
}

// dinv = (deg+1)^-1/2 ; invdeg = (deg+1)^-1
__global__ void dinv_kernel(const float* __restrict__ deg, float* __restrict__ dinv,
                            float* __restrict__ invdeg, int N) {
  int i = blockIdx.x * blockDim.x + threadIdx.x;
  if (i < N) {
    float d = deg[i] + 1.0f;
    dinv[i]   = rsqrtf(d);
    invdeg[i] = 1.0f / d;
  }
}

// h = x @ W  (N x 64)*(64 x 64), fp32 WMMA 16x16x4, K-loop of 16.
// Also initializes agg = h * invdeg[row] (self-loop term) so the scatter
// kernel only needs atomics.
// Block = 128 threads = 4 waves; wave w computes columns [16w,16w+16) of a
// 16-row strip. One wave owns one 16x16 C tile in 8 VGPRs.
//
// A-row handling: CLAMP (not predicate) keeps every load in-bounds with zero
// EXEC divergence. A clamped (duplicate) A row only affects its own C row,
// which the store guard drops, so results are unchanged for ragged N.
__global__ void __launch_bounds__(128)
gemm_selfloop_kernel(const float* __restrict__ x, const float* __restrict__ W,
                     const float* __restrict__ invdeg,
                     float* __restrict__ h, float* __restrict__ agg, int N) {
  const int wave    = threadIdx.x >> 5;   // 0..3 -> column tile
  const int lane    = threadIdx.x & 31;
  const int half    = lane >> 4;          // ISA 7.12.2 lane-half split
  const int l       = lane & 15;
  const int rowBase = blockIdx.x * 16;
  const int colBase = wave * 16;

  int arow = rowBase + l;                 // A-matrix: M = lane%16 (both halves)
  arow = (arow < N) ? arow : (N - 1);     // branch-free clamp
  // Lane's A stream: x[arow, kk*4 + 2*half + {0,1}]  -> one v2f load per step.
  const float* xr = x + (size_t)arow * FDIM + 2 * half;
  // Lane's B stream: W[kk*4 + 2*half + {0,1}, colBase + l].
  const float* wp = W + (size_t)(2 * half) * FDIM + colBase + l;

  v8f c = {};
  #pragma unroll
  for (int kk = 0; kk < 16; ++kk) {
    const v2f a = *(const v2f*)(xr + kk * 4);          // K pair (aligned b64)
    v2f b;
    b.x = wp[(size_t)kk * 4 * FDIM];                   // row K
    b.y = wp[(size_t)kk * 4 * FDIM + FDIM];            // row K+1
    c = __builtin_amdgcn_wmma_f32_16x16x4_f32(
        /*neg_a=*/false, a, /*neg_b=*/false, b,
        /*c_mod=*/(short)0, c, /*reuse_a=*/false, /*reuse_b=*/false);
  }

  // C/D layout: VGPR r -> M = r + 8*half, N = l.
  #pragma unroll
  for (int r = 0; r < 8; ++r) {
    const int row = rowBase + r + 8 * half;
    if (row < N) {
      const size_t idx = (size_t)row * FDIM + colBase + l;
      const float v = c[r];
      h[idx]   = v;
      agg[idx] = v * invdeg[row];
    }
  }
}

// Edge scatter: 16 lanes per edge, each lane moves a float4 (64 floats/edge).
// agg[dst] += dinv[src]*dinv[dst] * h[src]; atomics resolve in L2
// (h and agg are 25.6MB each -> resident in the 192MB L2).
__global__ void __launch_bounds__(256)
scatter_kernel(const int* __restrict__ ei, const float* __restrict__ dinv,
               const float* __restrict__ h, float* __restrict__ agg, int E) {
  const long gid = (long)blockIdx.x * blockDim.x + threadIdx.x;
  const int e = (int)(gid >> 4);
  const int q = (int)(gid & 15);
  if (e >= E) return;
  const int s = ei[e];
  const int d = ei[E + e];
  const float coef = dinv[s] * dinv[d];
  const float4 v = *((const float4*)(h + (size_t)s * FDIM) + q);
  float* ap = agg + (size_t)d * FDIM + (size_t)q * 4;
  unsafeAtomicAdd(ap + 0, v.x * coef);
  unsafeAtomicAdd(ap + 1, v.y * coef);
  unsafeAtomicAdd(ap + 2, v.z * coef);
  unsafeAtomicAdd(ap + 3, v.w * coef);
}

// out = (relu ? max(agg+b,0) : agg+b), float4 per thread.
__global__ void __launch_bounds__(256)
update_kernel(const float* __restrict__ agg, const float* __restrict__ b,
              float* __restrict__ out, int N, int do_relu) {
  const int gid = blockIdx.x * blockDim.x + threadIdx.x;  // N*16 float4 groups
  if (gid >= N * 16) return;
  const int q = gid & 15;                                  // column group
  float4 v  = ((const float4*)agg)[gid];
  const float4 bb = ((const float4*)b)[q];
  v.x += bb.x; v.y += bb.y; v.z += bb.z; v.w += bb.w;
  if (do_relu) {
    v.x = fmaxf(v.x, 0.0f); v.y = fmaxf(v.y, 0.0f);
    v.z = fmaxf(v.z, 0.0f); v.w = fmaxf(v.w, 0.0f);
  }
  ((float4*)out)[gid] = v;
}

extern "C" void kernel_launch(void* const* d_in, const int* in_sizes, int n_in,
                              void* d_out, int out_size, void* d_ws, size_t ws_size,
                              hipStream_t stream) {
  const float* x  = (const float*)d_in[0];   // [N,64] f32
  const int*   ei = (const int*)  d_in[1];   // [2,E]  i32 (row0=src, row1=dst)
  const float* W  = (const float*)d_in[2];   // [DEPTH,64,64] f32
  const float* b  = (const float*)d_in[3];   // [DEPTH,64] f32

  const int N     = in_sizes[0] / FDIM;
  const int E     = in_sizes[1] / 2;
  const int DEPTH = in_sizes[2] / (FDIM * FDIM);

  // Workspace layout (floats): deg[N] | dinv[N] | invdeg[N] | h[N*64] | agg[N*64] | xbuf[N*64]
  float* ws     = (float*)d_ws;
  float* deg    = ws;
  float* dinv   = ws + (size_t)N;
  float* invdeg = ws + (size_t)2 * N;
  float* h      = ws + (size_t)3 * N;
  float* agg    = h   + (size_t)N * FDIM;
  float* xbuf   = agg + (size_t)N * FDIM;

  const int T = 256;

  // Degrees (once; identical across layers).
  zero_f32_kernel<<<(N + T - 1) / T, T, 0, stream>>>(deg, N);
  degree_kernel<<<(E + T - 1) / T, T, 0, stream>>>(ei + E, deg, E);
  dinv_kernel<<<(N + T - 1) / T, T, 0, stream>>>(deg, dinv, invdeg, N);

  const int gemmBlocks    = (N + 15) / 16;
  const int scatterBlocks = (int)(((long)E * 16 + T - 1) / T);
  const int updateBlocks  = (N * 16 + T - 1) / T;

  const float* xcur = x;
  for (int layer = 0; layer < DEPTH; ++layer) {
    gemm_selfloop_kernel<<<gemmBlocks, 128, 0, stream>>>(
        xcur, W + (size_t)layer * FDIM * FDIM, invdeg, h, agg, N);
    scatter_kernel<<<scatterBlocks, T, 0, stream>>>(ei, dinv, h, agg, E);
    float* xout = (layer == DEPTH - 1) ? (float*)d_out : xbuf;
    update_kernel<<<updateBlocks, T, 0, stream>>>(
        agg, b + (size_t)layer * FDIM, xout, N, layer != DEPTH - 1);
    xcur = xout;
  }
}